// DarkChannelPriorLossV2_4148938407992
// MI455X (gfx1250) — compile-verified
//
#include <hip/hip_runtime.h>
#include <stdint.h>

// ---------------- Problem constants (reference shapes are fixed) ------------
#define NUM_DBINS 10
#define NGRAY     1024            // gray quantile histogram resolution
#define HWPIX     (1024 * 1024)   // H*W
#define PLANE4    (HWPIX / 4)     // float4 groups per plane = 2^18

// ---------------- Workspace layout (uint32 indices into d_ws) --------------
#define WS_MINBITS 0              // ordered-uint encoded global d min
#define WS_MAXBITS 1              // ordered-uint encoded global d max
#define WS_THR     8              // float[NUM_DBINS] thresholds
#define WS_HIST    32             // u32[NUM_DBINS*NGRAY]
#define WS_PART    (32 + NUM_DBINS * NGRAY)  // float[FINAL_GRID] partial sums

#define FINAL_GRID 1024
#define FINAL_BLK  256
#define TILE_G     256            // float4 groups per tile (== FINAL_BLK)
#define HIST_GRID  384

// ---------------- CDNA5 async global->LDS path (guarded) -------------------
#if defined(__HIP_DEVICE_COMPILE__) && defined(__gfx1250__) && \
    __has_builtin(__builtin_amdgcn_global_load_async_to_lds_b128) && \
    __has_builtin(__builtin_amdgcn_s_wait_asynccnt)
#define USE_ASYNC 1
#else
#define USE_ASYNC 0
#endif

#define AS1 __attribute__((address_space(1)))
#define AS3 __attribute__((address_space(3)))

typedef int v4i __attribute__((vector_size(16)));  // b128 payload type

// ---------------- Helpers ---------------------------------------------------
__device__ __forceinline__ uint32_t f2ord(float f) {
  uint32_t u = __float_as_uint(f);
  return (u & 0x80000000u) ? ~u : (u | 0x80000000u);
}
__device__ __forceinline__ float ord2f(uint32_t e) {
  uint32_t u = (e & 0x80000000u) ? (e & 0x7FFFFFFFu) : ~e;
  return __uint_as_float(u);
}
__device__ __forceinline__ float grayf(float r, float g, float b) {
  return fmaf(0.299f, r, fmaf(0.587f, g, 0.114f * b));
}
__device__ __forceinline__ float waveAdd(float v) {
  for (int off = 16; off > 0; off >>= 1) v += __shfl_xor(v, off, 32);
  return v;
}

// ---------------- Kernel 4 (listed first so disasm snippet shows async path):
// apply thresholds, write dcp, per-block partial L1 sums ----------------------
__global__ __launch_bounds__(FINAL_BLK) void k_final(const float4* __restrict__ rgb4,
                                                     const float4* __restrict__ d4,
                                                     uint32_t* ws,
                                                     float* __restrict__ out,
                                                     int n4) {
  __shared__ float4 sbuf[2][4][TILE_G];   // double-buffered r/g/b/d tiles, 32 KB
  __shared__ float  sthr[NUM_DBINS];
  __shared__ float  swsum[FINAL_BLK / 32];
  int tid = threadIdx.x;
  int ntiles = n4 / TILE_G;

  auto issue = [&](int tile, int bufsel) {
    int i   = tile * TILE_G + tid;
    int b   = i >> 18;
    int hw4 = i & (PLANE4 - 1);
    long rb = (long)b * 3 * PLANE4 + hw4;
#if USE_ASYNC
    __builtin_amdgcn_global_load_async_to_lds_b128(
        (AS1 v4i*)(rgb4 + rb),               (AS3 v4i*)&sbuf[bufsel][0][tid], 0, 0);
    __builtin_amdgcn_global_load_async_to_lds_b128(
        (AS1 v4i*)(rgb4 + rb + PLANE4),      (AS3 v4i*)&sbuf[bufsel][1][tid], 0, 0);
    __builtin_amdgcn_global_load_async_to_lds_b128(
        (AS1 v4i*)(rgb4 + rb + 2 * PLANE4),  (AS3 v4i*)&sbuf[bufsel][2][tid], 0, 0);
    __builtin_amdgcn_global_load_async_to_lds_b128(
        (AS1 v4i*)(d4 + i),                  (AS3 v4i*)&sbuf[bufsel][3][tid], 0, 0);
#else
    sbuf[bufsel][0][tid] = rgb4[rb];
    sbuf[bufsel][1][tid] = rgb4[rb + PLANE4];
    sbuf[bufsel][2][tid] = rgb4[rb + 2 * PLANE4];
    sbuf[bufsel][3][tid] = d4[i];
#endif
  };

  // Kick off tile-0 DMA immediately; scalar setup below overlaps with it.
  if ((int)blockIdx.x < ntiles) issue((int)blockIdx.x, 0);

  if (tid < NUM_DBINS) sthr[tid] = ((const float*)ws)[WS_THR + tid];
  __syncthreads();

  float dmin = ord2f(ws[WS_MINBITS]);
  float dmax = ord2f(ws[WS_MAXBITS]);
  float range = dmax - dmin;
  float bs = (range > 0.f) ? ((float)NUM_DBINS / range) : 0.f;

  auto dcp1 = [&](float r_, float g_, float b_, float d_) -> float {
    float gy = grayf(r_, g_, b_);
    float t = (d_ - dmin) * bs;
    if (bs > 0.f && t >= 0.f && t < (float)NUM_DBINS) {
      int db = (int)t;
      if (gy <= sthr[db]) return gy;
    }
    return 0.f;
  };

  float lsum = 0.f;
  int bb = 0;
  for (int tile = (int)blockIdx.x; tile < ntiles; tile += (int)gridDim.x) {
    int nxt = tile + (int)gridDim.x;
    if (nxt < ntiles) issue(nxt, bb ^ 1);
#if USE_ASYNC
    if (nxt < ntiles) __builtin_amdgcn_s_wait_asynccnt(4);  // current tile done
    else              __builtin_amdgcn_s_wait_asynccnt(0);
#endif
    float4 r  = sbuf[bb][0][tid];
    float4 g  = sbuf[bb][1][tid];
    float4 bl = sbuf[bb][2][tid];
    float4 dd = sbuf[bb][3][tid];
    float o0 = dcp1(r.x, g.x, bl.x, dd.x);
    float o1 = dcp1(r.y, g.y, bl.y, dd.y);
    float o2 = dcp1(r.z, g.z, bl.z, dd.z);
    float o3 = dcp1(r.w, g.w, bl.w, dd.w);
    long i  = (long)tile * TILE_G + tid;
    long ob = 1 + 4 * i;                 // out[0] is the loss scalar
    out[ob + 0] = o0; out[ob + 1] = o1; out[ob + 2] = o2; out[ob + 3] = o3;
    lsum += fabsf(o0) + fabsf(o1) + fabsf(o2) + fabsf(o3);
    bb ^= 1;
  }

  lsum = waveAdd(lsum);
  int wid = tid >> 5, lid = tid & 31;
  if (lid == 0) swsum[wid] = lsum;
  __syncthreads();
  if (tid == 0) {
    float s = 0.f;
    for (int w = 0; w < FINAL_BLK / 32; ++w) s += swsum[w];
    ((float*)(ws + WS_PART))[blockIdx.x] = s;   // deterministic: one slot/block
  }
}

// ---------------- Kernel 0: init workspace ----------------------------------
__global__ void k_init(uint32_t* ws) {
  int i = blockIdx.x * blockDim.x + threadIdx.x;
  if (i == 0) {
    ws[WS_MINBITS] = 0xFFFFFFFFu;
    ws[WS_MAXBITS] = 0u;
  }
  int total = WS_PART + FINAL_GRID;
  for (int j = i; j < total; j += gridDim.x * blockDim.x)
    if (j >= WS_THR) ws[j] = 0u;
}

// ---------------- Kernel 1: global min/max of d -----------------------------
__global__ __launch_bounds__(256) void k_minmax(const float4* __restrict__ d4,
                                                uint32_t* ws, int n4) {
  int stride = gridDim.x * blockDim.x;
  float lmin = __FLT_MAX__, lmax = -__FLT_MAX__;
  for (int i = blockIdx.x * blockDim.x + threadIdx.x; i < n4; i += stride) {
    __builtin_prefetch(&d4[i + stride], 0, 0);  // gfx1250 global_prefetch
    float4 v = d4[i];
    lmin = fminf(lmin, fminf(fminf(v.x, v.y), fminf(v.z, v.w)));
    lmax = fmaxf(lmax, fmaxf(fmaxf(v.x, v.y), fmaxf(v.z, v.w)));
  }
  for (int off = 16; off > 0; off >>= 1) {
    lmin = fminf(lmin, __shfl_xor(lmin, off, 32));
    lmax = fmaxf(lmax, __shfl_xor(lmax, off, 32));
  }
  __shared__ float smin[8], smax[8];
  int wid = threadIdx.x >> 5, lid = threadIdx.x & 31;
  if (lid == 0) { smin[wid] = lmin; smax[wid] = lmax; }
  __syncthreads();
  if (threadIdx.x == 0) {
    float m = smin[0], M = smax[0];
    for (int w = 1; w < 8; ++w) { m = fminf(m, smin[w]); M = fmaxf(M, smax[w]); }
    atomicMin(&ws[WS_MINBITS], f2ord(m));
    atomicMax(&ws[WS_MAXBITS], f2ord(M));
  }
}

// ---------------- Kernel 2: (depth-bin x gray-bucket) histogram --------------
__global__ __launch_bounds__(256) void k_hist(const float4* __restrict__ rgb4,
                                              const float4* __restrict__ d4,
                                              uint32_t* ws, int n4) {
  __shared__ uint32_t lhist[NUM_DBINS * NGRAY];  // 40 KB LDS
  for (int j = threadIdx.x; j < NUM_DBINS * NGRAY; j += blockDim.x) lhist[j] = 0u;
  __syncthreads();

  float dmin = ord2f(ws[WS_MINBITS]);
  float dmax = ord2f(ws[WS_MAXBITS]);
  float range = dmax - dmin;
  float bs = (range > 0.f) ? ((float)NUM_DBINS / range) : 0.f;

  int stride = gridDim.x * blockDim.x;
  for (int i = blockIdx.x * blockDim.x + threadIdx.x; i < n4; i += stride) {
    int b   = i >> 18;                 // batch index (pixel>>20)
    int hw4 = i & (PLANE4 - 1);
    long rb = (long)b * 3 * PLANE4 + hw4;
    float4 r  = rgb4[rb];
    float4 g  = rgb4[rb + PLANE4];
    float4 bl = rgb4[rb + 2 * PLANE4];
    float4 dd = d4[i];
    float rr[4] = {r.x, r.y, r.z, r.w};
    float gg[4] = {g.x, g.y, g.z, g.w};
    float bb[4] = {bl.x, bl.y, bl.z, bl.w};
    float dv[4] = {dd.x, dd.y, dd.z, dd.w};
#pragma unroll
    for (int c = 0; c < 4; ++c) {
      float t = (dv[c] - dmin) * bs;
      if (bs > 0.f && t >= 0.f && t < (float)NUM_DBINS) {
        int db = (int)t;
        float gy = grayf(rr[c], gg[c], bb[c]);
        int gq = (int)(gy * (float)NGRAY);
        gq = gq < 0 ? 0 : (gq > NGRAY - 1 ? NGRAY - 1 : gq);
        atomicAdd(&lhist[db * NGRAY + gq], 1u);
      }
    }
  }
  __syncthreads();
  for (int j = threadIdx.x; j < NUM_DBINS * NGRAY; j += blockDim.x) {
    uint32_t c = lhist[j];
    if (c) atomicAdd(&ws[WS_HIST + j], c);
  }
}

// ---------------- Kernel 3: per-bin 1% quantile threshold --------------------
__global__ void k_select(uint32_t* ws) {
  int b = threadIdx.x;
  if (b >= NUM_DBINS) return;
  const uint32_t* h = ws + WS_HIST + b * NGRAY;
  uint32_t n = 0;
  for (int j = 0; j < NGRAY; ++j) n += h[j];
  float* thr = (float*)ws + WS_THR;
  if (n == 0) { thr[b] = -1.0f; return; }   // empty bin: mask is empty anyway
  uint32_t k = (uint32_t)ceilf((float)n * 0.01f);
  if (k < 1u) k = 1u;
  uint32_t c = 0; int j = 0;
  for (; j < NGRAY; ++j) { c += h[j]; if (c >= k) break; }
  thr[b] = (float)(j + 1) / (float)NGRAY;   // upper edge of k-th bucket
}

// ---------------- Kernel 5: finalize loss ------------------------------------
__global__ void k_loss(const float* __restrict__ part, float* __restrict__ out,
                       int np, float invn) {
  __shared__ float sw[8];
  float v = 0.f;
  for (int i = threadIdx.x; i < np; i += blockDim.x) v += part[i];
  v = waveAdd(v);
  int wid = threadIdx.x >> 5, lid = threadIdx.x & 31;
  if (lid == 0) sw[wid] = v;
  __syncthreads();
  if (threadIdx.x == 0) {
    float s = 0.f;
    for (int w = 0; w < 8; ++w) s += sw[w];
    out[0] = s * invn;
  }
}

// ---------------- Launch -----------------------------------------------------
extern "C" void kernel_launch(void* const* d_in, const int* in_sizes, int n_in,
                              void* d_out, int out_size, void* d_ws, size_t ws_size,
                              hipStream_t stream) {
  const float* rgb = (const float*)d_in[0];
  const float* dep = (const float*)d_in[1];
  float* out = (float*)d_out;
  uint32_t* ws = (uint32_t*)d_ws;
  int n  = in_sizes[1];   // 8*1024*1024 depth pixels
  int n4 = n >> 2;

  k_init  <<<64,   256, 0, stream>>>(ws);
  k_minmax<<<1024, 256, 0, stream>>>((const float4*)dep, ws, n4);
  k_hist  <<<HIST_GRID, 256, 0, stream>>>((const float4*)rgb, (const float4*)dep,
                                          ws, n4);
  k_select<<<1,    32,  0, stream>>>(ws);
  k_final <<<FINAL_GRID, FINAL_BLK, 0, stream>>>((const float4*)rgb,
                                                 (const float4*)dep, ws, out, n4);
  k_loss  <<<1, 256, 0, stream>>>((const float*)(ws + WS_PART), out, FINAL_GRID,
                                  1.0f / (float)n);
}